// LinearCRF_74466142978185
// MI455X (gfx1250) — compile-verified
//
#include <hip/hip_runtime.h>
#include <stdint.h>

// Viterbi CRF decode for MI455X (gfx1250, wave32).
// B=512 sequences, L=2048, K=19 tags. One wave per sequence (512 waves):
// ~1 wave/SIMD -> the kernel is dependence-chain bound, so the K-argmax is
// split into 4 independent compare/select chains (exact first-max tie-break
// preserved by contiguous ranges + earlier-range-wins merge).
// - Emissions streamed global->LDS with GLOBAL_LOAD_ASYNC_TO_LDS_B128
//   (ASYNCcnt) double-buffering; marked TH_LOAD_NT: P is 80 MB streamed once,
//   keep the 192 MB L2 for the 20 MB backpointer array instead.
// - Backpointers stored transposed (u8 [B][K][L]) in d_ws, L2-resident.
// - Backtrace: 32-step chunks preloaded to registers, dependent chain is
//   v_readlane + byte-extract only (no dependent memory loads).
// WMMA not applicable: (max,+) semiring with argmax, K=19.

#define NB       512
#define SEQL     2048
#define NK       19
#define START_ID 17
#define PAD_ID   18
#define CT       64                    // timesteps per LDS chunk
#define CHUNK_BYTES (CT * NK * 4)      // 4864 bytes
#define WPB      4                     // waves per block

__global__ __launch_bounds__(WPB * 32, 1)
void viterbi_fused(const float* __restrict__ P,
                   const float* __restrict__ T,
                   const int*   __restrict__ mask,
                   int*         __restrict__ out,
                   unsigned char* __restrict__ bp /* [NB][NK][SEQL] */)
{
    __shared__ float ebuf[WPB][2][CT * NK];   // 2 x 4864 B per wave

    const int lane = threadIdx.x & 31;
    const int w    = threadIdx.x >> 5;
    const int b    = blockIdx.x * WPB + w;
    if (b >= NB) return;

    // ---- sequence length = popcount of contiguous-prefix mask row ----
    const int* mrow = mask + (size_t)b * SEQL;
    int len = 0;
    for (int t = lane; t < SEQL; t += 32) len += (__builtin_nontemporal_load(mrow + t) != 0);
    #pragma unroll
    for (int off = 16; off; off >>= 1) len += __shfl_xor(len, off, 32);
    // len is wave-uniform, in [1, SEQL]

    // ---- transition row for this lane's tag: Trow[k] = T[j][k] ----
    const int j = (lane < NK) ? lane : 0;
    float Trow[NK];
    #pragma unroll
    for (int k = 0; k < NK; ++k) Trow[k] = T[j * NK + k];

    float DP = (lane == START_ID) ? 0.0f : -1000.0f;

    const float*   prow = P + (size_t)b * SEQL * NK;
    const uint32_t lds0 = (uint32_t)(uintptr_t)&ebuf[w][0][0];
    const uint32_t lds1 = (uint32_t)(uintptr_t)&ebuf[w][1][0];
    const int nchunks   = (len + CT - 1) / CT;

    // ---- kick off async DMA of chunk 0 into LDS buffer 0 ----
    {
        uint64_t g0 = (uint64_t)(uintptr_t)prow;
        for (int ofs = lane * 16; ofs < CHUNK_BYTES; ofs += 512) {
            uint32_t l = lds0 + (uint32_t)ofs;
            uint64_t g = g0 + (uint64_t)ofs;
            asm volatile("global_load_async_to_lds_b128 %0, %1, off th:TH_LOAD_NT"
                         :: "v"(l), "v"(g) : "memory");
        }
    }

    unsigned char* bprow = bp + ((size_t)b * NK + (size_t)j) * SEQL;

    // broadcast DP[k] as wave-uniform SGPR
    auto rd = [&](int k) -> float {
        return __int_as_float(__builtin_amdgcn_readlane(__float_as_int(DP), k));
    };

    // ---- forward recursion with double-buffered emissions ----
    for (int c = 0; c < nchunks; ++c) {
        asm volatile("s_wait_asynccnt 0x0" ::: "memory");   // chunk c resident

        if (c + 1 < nchunks) {                               // prefetch chunk c+1
            uint64_t g0 = (uint64_t)(uintptr_t)(prow + (size_t)(c + 1) * CT * NK);
            uint32_t lb = ((c + 1) & 1) ? lds1 : lds0;
            for (int ofs = lane * 16; ofs < CHUNK_BYTES; ofs += 512) {
                uint32_t l = lb + (uint32_t)ofs;
                uint64_t g = g0 + (uint64_t)ofs;
                asm volatile("global_load_async_to_lds_b128 %0, %1, off th:TH_LOAD_NT"
                             :: "v"(l), "v"(g) : "memory");
            }
        }

        const float* eb   = &ebuf[w][c & 1][0];
        const int    tend = min(CT, len - c * CT);
        for (int tl = 0; tl < tend; ++tl) {
            float emit = eb[tl * NK + j];

            // 4 independent compare/select chains over contiguous k ranges:
            // [0,5) [5,10) [10,15) [15,19). Earlier-range-wins merge keeps
            // exact first-occurrence argmax semantics.
            float m0 = rd(0)  + Trow[0];  int a0 = 0;
            float m1 = rd(5)  + Trow[5];  int a1 = 5;
            float m2 = rd(10) + Trow[10]; int a2 = 10;
            float m3 = rd(15) + Trow[15]; int a3 = 15;
            #pragma unroll
            for (int k = 1; k < 5; ++k) {
                float c0 = rd(k)      + Trow[k];
                float c1 = rd(k + 5)  + Trow[k + 5];
                float c2 = rd(k + 10) + Trow[k + 10];
                if (c0 > m0) { m0 = c0; a0 = k; }
                if (c1 > m1) { m1 = c1; a1 = k + 5; }
                if (c2 > m2) { m2 = c2; a2 = k + 10; }
                if (k < 4) {
                    float c3 = rd(k + 15) + Trow[k + 15];
                    if (c3 > m3) { m3 = c3; a3 = k + 15; }
                }
            }
            if (m1 > m0) { m0 = m1; a0 = a1; }
            if (m3 > m2) { m2 = m3; a2 = a3; }
            if (m2 > m0) { m0 = m2; a0 = a2; }

            float nDP = m0 + emit;
            if (lane < NK) {
                DP = nDP;
                bprow[c * CT + tl] = (unsigned char)a0;   // RT: keep in L2
            }
        }
    }

    // ---- transition into PAD, argmax -> last tag (uniform) ----
    float fin = (lane < NK) ? (DP + T[PAD_ID * NK + j]) : -3.0e38f;
    int last = 0; float bm = -3.0e38f;
    #pragma unroll
    for (int k = 0; k < NK; ++k) {
        float v = __int_as_float(__builtin_amdgcn_readlane(__float_as_int(fin), k));
        if (v > bm) { bm = v; last = k; }
    }

    int* orow = out + (size_t)b * SEQL;
    for (int t = len + lane; t < SEQL; t += 32)
        __builtin_nontemporal_store(-1, orow + t);          // pad tail

    // our own backpointer stores must be visible to our loads
    asm volatile("s_wait_storecnt 0x0" ::: "memory");

    // ---- backtrace: 32-step register chunks, readlane-only chain ----
    int cur = last;                                   // wave-uniform
    const unsigned char* bpb = bp + (size_t)b * NK * SEQL;
    for (int t0 = ((len - 1) >> 5) << 5; t0 >= 0; t0 -= 32) {
        uint32_t wd[8];
        if (lane < NK) {
            const uint4* p4 = (const uint4*)(bpb + (size_t)lane * SEQL + t0);
            uint4 q = p4[0], r = p4[1];               // 32 bytes of row `lane`
            wd[0] = q.x; wd[1] = q.y; wd[2] = q.z; wd[3] = q.w;
            wd[4] = r.x; wd[5] = r.y; wd[6] = r.z; wd[7] = r.w;
        } else {
            #pragma unroll
            for (int i = 0; i < 8; ++i) wd[i] = 0;
        }
        #pragma unroll
        for (int i = 31; i >= 0; --i) {
            int t = t0 + i;
            if (t >= 1 && t < len) {                  // uniform branch
                if (lane == 0) __builtin_nontemporal_store(cur, orow + t);
                uint32_t word = (uint32_t)__builtin_amdgcn_readlane(
                    (int)wd[i >> 2], cur);            // choice[t][cur] lane select
                cur = (int)((word >> ((i & 3) * 8)) & 31u);
            }
        }
    }
    if (lane == 0) __builtin_nontemporal_store(cur, orow + 0);
}

extern "C" void kernel_launch(void* const* d_in, const int* in_sizes, int n_in,
                              void* d_out, int out_size, void* d_ws, size_t ws_size,
                              hipStream_t stream) {
    const float* P    = (const float*)d_in[0];   // [512, 2048, 19] f32
    const float* T    = (const float*)d_in[1];   // [19, 19] f32
    const int*   mask = (const int*)d_in[2];     // [512, 2048] int (bool)
    int* out = (int*)d_out;                      // [512, 2048] int32 path
    unsigned char* bp = (unsigned char*)d_ws;    // needs NB*NK*SEQL = 19,922,944 B
    (void)in_sizes; (void)n_in; (void)out_size; (void)ws_size;

    dim3 grid(NB / WPB);          // 128 blocks
    dim3 block(WPB * 32);         // 4 waves/block, 1 sequence per wave
    viterbi_fused<<<grid, block, 0, stream>>>(P, T, mask, out, bp);
}